// GraphConvolutionLayer_4776003633738
// MI455X (gfx1250) — compile-verified
//
#include <hip/hip_runtime.h>
#include <hip/hip_bf16.h>

#define N_NODES  100000
#define N_EDGES  1600000
#define IN_F     128   // K
#define OUT_F    64    // N

typedef __attribute__((ext_vector_type(2))) float v2f;
typedef __attribute__((ext_vector_type(8))) float v8f;

// -------------------------------------------------------------------------
// Kernel 1: support = x @ W  via V_WMMA_F32_16X16X4_F32 (fp32-exact path).
// Block = 256 threads = 8 waves: wave>>2 selects M-tile (2), wave&3 selects
// N-tile (4). Block covers 32 rows x 64 cols; grid = 100000/32 = 3125.
// B fragments (W is 128x64 = 32KB, L2/L1 resident) preloaded into 64 VGPRs
// per wave and reused across all 32 K-steps.
// -------------------------------------------------------------------------
__global__ __launch_bounds__(256) void gcn_gemm_wmma(const float* __restrict__ x,
                                                     const float* __restrict__ w,
                                                     float* __restrict__ support) {
    const int lane = threadIdx.x & 31;
    const int wave = threadIdx.x >> 5;
    const int m_base = blockIdx.x * 32 + (wave >> 2) * 16;
    const int n_base = (wave & 3) * 16;
    const int l15 = lane & 15;
    const int kk  = (lane >> 4) * 2;       // lanes 0-15 -> K {0,1}; 16-31 -> K {2,3}

    const int row = m_base + l15;          // A fragment row (M)
    const int n   = n_base + l15;          // B fragment col (N)

    // Preload all B fragments: b[i] covers K = 4*i + {kk, kk+1} at column n.
    v2f bfrag[32];
#pragma unroll
    for (int i = 0; i < 32; ++i) {
        const int k = i * 4 + kk;
        bfrag[i][0] = w[k * OUT_F + n];
        bfrag[i][1] = w[(k + 1) * OUT_F + n];
    }

    const float* xrow = x + (size_t)row * IN_F + kk;
    v8f acc = {0.f, 0.f, 0.f, 0.f, 0.f, 0.f, 0.f, 0.f};
#pragma unroll
    for (int i = 0; i < 32; ++i) {
        v2f a;
        a[0] = xrow[i * 4];
        a[1] = xrow[i * 4 + 1];
        // 8 args: (neg_a, A, neg_b, B, c_mod, C, reuse_a, reuse_b)
        acc = __builtin_amdgcn_wmma_f32_16x16x4_f32(
            false, a, false, bfrag[i], (short)0, acc, false, false);
    }

    // C/D layout: VGPR r -> M = r (lanes 0-15) or r+8 (lanes 16-31), N = lane&15.
    const int cm = m_base + (lane >> 4) * 8;
    float* outp = support + (size_t)cm * OUT_F + n_base + l15;
#pragma unroll
    for (int r = 0; r < 8; ++r) outp[(size_t)r * OUT_F] = acc[r];
}

// -------------------------------------------------------------------------
// Kernel 2: out[n][c] = bias[c]  (bias folded in before scatter; also clears
// the 0xAA poison the harness leaves in d_out).
// -------------------------------------------------------------------------
__global__ void gcn_init_bias(const float* __restrict__ bias, float* __restrict__ out) {
    const int i = blockIdx.x * blockDim.x + threadIdx.x;
    if (i < N_NODES * OUT_F) out[i] = bias[i & (OUT_F - 1)];
}

// -------------------------------------------------------------------------
// Kernel 3: edge scatter. 16 lanes per edge; each lane gathers a float4 of
// support[src] (global_load_b128, L2-resident) and issues 4 native fp32
// atomics (global_atomic_add_f32 via unsafeAtomicAdd) into out[dst].
// -------------------------------------------------------------------------
__global__ __launch_bounds__(256) void gcn_scatter(const float* __restrict__ support,
                                                   const int*   __restrict__ esrc,
                                                   const int*   __restrict__ edst,
                                                   const float* __restrict__ evals,
                                                   float* __restrict__ out) {
    const int gid = blockIdx.x * blockDim.x + threadIdx.x;   // 25.6M < 2^31
    const int e = gid >> 4;
    if (e >= N_EDGES) return;
    const int c = (gid & 15) * 4;

    const int   s = esrc[e];
    const int   d = edst[e];
    const float v = evals[e];

    const float4 sv = *(const float4*)(support + (size_t)s * OUT_F + c);
    float* o = out + (size_t)d * OUT_F + c;
    unsafeAtomicAdd(o + 0, v * sv.x);
    unsafeAtomicAdd(o + 1, v * sv.y);
    unsafeAtomicAdd(o + 2, v * sv.z);
    unsafeAtomicAdd(o + 3, v * sv.w);
}

// -------------------------------------------------------------------------
extern "C" void kernel_launch(void* const* d_in, const int* in_sizes, int n_in,
                              void* d_out, int out_size, void* d_ws, size_t ws_size,
                              hipStream_t stream) {
    const float* x      = (const float*)d_in[0];   // [100000,128]
    const float* weight = (const float*)d_in[1];   // [128,64]
    const float* bias   = (const float*)d_in[2];   // [64]
    const int*   esrc   = (const int*)  d_in[3];   // [1.6M]
    const int*   edst   = (const int*)  d_in[4];   // [1.6M]
    const float* evals  = (const float*)d_in[5];   // [1.6M]
    float* out = (float*)d_out;                    // [100000,64]

    float* support = (float*)d_ws;                 // 100000*64*4 = 25.6 MB scratch

    // 1) support = x @ W  (WMMA fp32)
    gcn_gemm_wmma<<<N_NODES / 32, 256, 0, stream>>>(x, weight, support);

    // 2) out = bias (broadcast)
    const int total = N_NODES * OUT_F;
    gcn_init_bias<<<(total + 255) / 256, 256, 0, stream>>>(bias, out);

    // 3) out[dst] += val * support[src]
    const long long work = (long long)N_EDGES * 16;
    gcn_scatter<<<(int)(work / 256), 256, 0, stream>>>(support, esrc, edst, evals, out);
}